// VectorQuantizer_38714835206744
// MI455X (gfx1250) — compile-verified
//
#include <hip/hip_runtime.h>
#include <hip/hip_bf16.h>

typedef float v2f __attribute__((ext_vector_type(2)));
typedef float v8f __attribute__((ext_vector_type(8)));

#define HWSZ 4096          // H*W
#define DSZ  64            // embedding dim
#define KSZ  1024          // codebook size
#define NSZ  65536         // B*H*W vectors
#define ZQ_ELEMS 4194304   // B*D*H*W

// ---------------------------------------------------------------------------
// Kernel 1: repack embedding into 32-code chunks as contiguous d-pairs with a
// bank-swizzle:
//   p = d/2, j = d%2, n' = n ^ ((p&1)<<4)
//   ETp[c*2048 + (p*32 + n')*2 + j] = E[c*32 + n][d]
// So in the GEMM kernel, one B fragment (d0,d0+1 for one code) is a single
// aligned ds_load_b64, and the two lane-halves (p parity differs) plus the two
// code tiles hit disjoint LDS bank halves. Also compute hn[k] = 0.5*||E[k]||^2
// and zero the loss accumulator.
// ---------------------------------------------------------------------------
__global__ __launch_bounds__(256) void vq_prep_kernel(
    const float* __restrict__ E, float* __restrict__ ETp,
    float* __restrict__ hn, float* __restrict__ loss)
{
    int k = blockIdx.x * 256 + threadIdx.x;   // 0..1023
    if (k == 0) *loss = 0.0f;
    if (k >= KSZ) return;
    int c  = k >> 5;      // chunk of 32 codes
    int nl = k & 31;      // code within chunk
    const float* e = E + (size_t)k * DSZ;
    float s = 0.0f;
    #pragma unroll
    for (int d = 0; d < DSZ; ++d) {
        float v = e[d];
        s += v * v;
        int p  = d >> 1;
        int j  = d & 1;
        int nn = nl ^ ((p & 1) << 4);                 // bank swizzle
        ETp[c * 2048 + (p * 32 + nn) * 2 + j] = v;
    }
    hn[k] = 0.5f * s;
}

// ---------------------------------------------------------------------------
// Kernel 2: fused f32 WMMA GEMM + argmin.
// One wave owns 16 rows of flat z. Per iteration: 32 codes as two 16-wide
// N-tiles -> two independent chains of 16x v_wmma_f32_16x16x4_f32 sharing the
// same A fragments. LDS codebook tile is double-buffered; global prefetch of
// tile c+1 is issued before the barrier so it overlaps the WMMAs of tile c.
// B fragments are single ds_load_b64 (pre-swizzled pair layout, conflict-free).
//
// f32 WMMA 16x16x4 fragment layout (wave32):
//   A (16x4):  lanes 0-15 M=0..15; VGPR0 = K d0 (lanes<16) / d0+2 (lanes>=16)
//   B (4x16):  lanes 0-15 N=0..15; same K split across lane halves
//   C (16x16): lanes 0-15 N=lane, VGPR r = M=r; lanes 16-31 N=lane-16, M=8+r
// ---------------------------------------------------------------------------
__global__ __launch_bounds__(256) void vq_argmin_kernel(
    const float* __restrict__ z, const float* __restrict__ ETp,
    const float* __restrict__ hn, int* __restrict__ idx_i,
    float* __restrict__ idx_f)
{
    __shared__ float sB[2][DSZ * 32];   // double-buffered 64x32 tile (2x8 KB)
    __shared__ float sHn[KSZ];          // half norms (4 KB)

    const int tid  = threadIdx.x;
    const int wave = tid >> 5;
    const int lane = tid & 31;
    const int hf   = lane >> 4;      // half-wave select (K split; == p parity)
    const int mrow = lane & 15;      // M row (for A) / N col (for B,C)

    for (int i = tid; i < KSZ; i += 256) sHn[i] = hn[i];

    const int tile   = blockIdx.x * 8 + wave;   // 16-row tile id
    const int m_base = tile * 16;
    const int bimg   = m_base >> 12;            // / 4096
    const int hw0    = m_base & 4095;

    // Preload A fragments: z[b][d][hw0+mrow], d split per WMMA layout.
    const float* zb = z + (size_t)bimg * DSZ * HWSZ + hw0 + mrow;
    v2f A[16];
    #pragma unroll
    for (int s = 0; s < 16; ++s) {
        int d0 = 4 * s + 2 * hf;
        A[s].x = zb[(size_t)d0 * HWSZ];
        A[s].y = zb[(size_t)(d0 + 1) * HWSZ];
    }

    float best[8];
    int   bidx[8];
    #pragma unroll
    for (int r = 0; r < 8; ++r) { best[r] = -3.4e38f; bidx[r] = 0; }

    // Swizzled n positions for this lane (p parity == hf for all its loads).
    const int n0 = mrow ^ (hf << 4);   // code tile 0
    const int n1 = n0 ^ 16;            // code tile 1

    // Prologue: stage chunk 0 (2048 floats = 2 float4 per thread).
    const float4* gsrc = reinterpret_cast<const float4*>(ETp);
    float4 nxt0 = gsrc[tid];
    float4 nxt1 = gsrc[256 + tid];
    reinterpret_cast<float4*>(&sB[0][0])[tid]       = nxt0;
    reinterpret_cast<float4*>(&sB[0][0])[256 + tid] = nxt1;

    for (int c = 0; c < 32; ++c) {
        const int buf = c & 1;
        if (c + 1 < 32) {   // prefetch next tile into registers
            nxt0 = gsrc[(c + 1) * 512 + tid];
            nxt1 = gsrc[(c + 1) * 512 + 256 + tid];
        }
        __syncthreads();    // sB[buf] ready; prior readers of sB[buf^1] done

        const v2f* sbp = reinterpret_cast<const v2f*>(&sB[buf][0]);
        v8f C0 = {};
        v8f C1 = {};
        #pragma unroll
        for (int s = 0; s < 16; ++s) {
            int p0 = 2 * s + hf;                 // d-pair index for this lane
            v2f B0 = sbp[p0 * 32 + n0];          // one ds_load_b64 each
            v2f B1 = sbp[p0 * 32 + n1];
            C0 = __builtin_amdgcn_wmma_f32_16x16x4_f32(
                     false, A[s], false, B0, (short)0, C0, false, false);
            C1 = __builtin_amdgcn_wmma_f32_16x16x4_f32(
                     false, A[s], false, B1, (short)0, C1, false, false);
        }

        // Per-lane codes are strictly increasing across iterations, so a
        // strict > keeps the first (lowest) index on ties — argmin semantics.
        const int code0 = c * 32 + mrow;
        const int code1 = code0 + 16;
        const float h0 = sHn[code0];
        const float h1 = sHn[code1];
        #pragma unroll
        for (int r = 0; r < 8; ++r) {
            float v0 = C0[r] - h0;   // maximize z.e - 0.5||e||^2
            if (v0 > best[r]) { best[r] = v0; bidx[r] = code0; }
            float v1 = C1[r] - h1;
            if (v1 > best[r]) { best[r] = v1; bidx[r] = code1; }
        }

        if (c + 1 < 32) {   // store prefetched tile into the other buffer
            float4* dst = reinterpret_cast<float4*>(&sB[buf ^ 1][0]);
            dst[tid]       = nxt0;
            dst[256 + tid] = nxt1;
        }
    }

    // Reduce (value, index) across the 16 lanes of each half-wave;
    // indices are unordered across lanes, so tie-break on smaller index.
    #pragma unroll
    for (int off = 8; off >= 1; off >>= 1) {
        #pragma unroll
        for (int r = 0; r < 8; ++r) {
            float ov = __shfl_xor(best[r], off, 16);
            int   oi = __shfl_xor(bidx[r], off, 16);
            if (ov > best[r] || (ov == best[r] && oi < bidx[r])) {
                best[r] = ov; bidx[r] = oi;
            }
        }
    }

    // lane 0 holds rows m_base+0..7, lane 16 holds rows m_base+8..15
    if (mrow == 0) {
        int rbase = m_base + hf * 8;
        #pragma unroll
        for (int r = 0; r < 8; ++r) {
            idx_i[rbase + r] = bidx[r];
            idx_f[rbase + r] = (float)bidx[r];
        }
    }
}

// ---------------------------------------------------------------------------
// Kernel 3: gather z_q = E[idx], write z_q_st = z_e + (z_q - z_e) in (B,D,H,W)
// layout, and accumulate vq_loss = (1 + BETA) * mean((z_q - z_e)^2).
// ---------------------------------------------------------------------------
__global__ __launch_bounds__(256) void vq_output_kernel(
    const float* __restrict__ z, const float* __restrict__ E,
    const int* __restrict__ idx_i, float* __restrict__ zq_out,
    float* __restrict__ loss)
{
    const int n    = blockIdx.x * 256 + threadIdx.x;   // 0..65535
    const int bimg = n >> 12;
    const int hw   = n & 4095;
    const int code = idx_i[n];
    const float* e  = E + (size_t)code * DSZ;
    const float* zp = z + (size_t)bimg * DSZ * HWSZ + hw;
    float*       op = zq_out + (size_t)bimg * DSZ * HWSZ + hw;

    float acc = 0.0f;
    #pragma unroll
    for (int d = 0; d < DSZ; ++d) {
        float q  = e[d];
        float ze = zp[(size_t)d * HWSZ];
        float df = q - ze;
        op[(size_t)d * HWSZ] = ze + df;   // straight-through value, bit-faithful
        acc += df * df;
    }
    // loss = (codebook + 0.25*commit) = 1.25 * mean(df^2)
    acc *= 1.25f / (float)ZQ_ELEMS;
    #pragma unroll
    for (int off = 16; off >= 1; off >>= 1) acc += __shfl_xor(acc, off, 32);
    if ((threadIdx.x & 31) == 0) atomicAdd(loss, acc);
}

// ---------------------------------------------------------------------------
extern "C" void kernel_launch(void* const* d_in, const int* in_sizes, int n_in,
                              void* d_out, int out_size, void* d_ws, size_t ws_size,
                              hipStream_t stream)
{
    const float* z_e = (const float*)d_in[0];   // (16,64,64,64)
    const float* E   = (const float*)d_in[1];   // (1024,64)

    // Workspace layout (floats): ETp[65536] | hn[1024] | idx_i[65536 ints]
    float* ETp  = (float*)d_ws;
    float* hn   = ETp + 65536;
    int*   idxi = (int*)(hn + KSZ);

    // Output layout: z_q_st[4194304] | vq_loss[1] | indices(float)[65536]
    float* zq   = (float*)d_out;
    float* loss = zq + ZQ_ELEMS;
    float* idxf = loss + 1;

    vq_prep_kernel<<<4, 256, 0, stream>>>(E, ETp, hn, loss);
    vq_argmin_kernel<<<512, 256, 0, stream>>>(z_e, ETp, hn, idxi, idxf);
    vq_output_kernel<<<256, 256, 0, stream>>>(z_e, E, idxi, zq, loss);
}